// STFT_60017872995039
// MI455X (gfx1250) — compile-verified
//
#include <hip/hip_runtime.h>
#include <hip/hip_bf16.h>
#include <math.h>

// ---------------- problem constants ----------------
#define N_FFT     1024
#define HOP       256
#define WIN       1024
#define SIG_LEN   16777216            // 2^24
#define PAD       768                 // 3*HOP zero pad each side
#define TOTAL     (SIG_LEN + 2*PAD)   // 16778752
#define N_ROWS    (TOTAL / HOP)       // 65542
#define N_FRAMES  ((TOTAL - WIN)/HOP + 1)  // 65539
#define HALF_N    (N_FFT/2 + 1)       // 513
#define NCOL_PAD  576                 // 36 bin-tiles of 16 (9 groups x 4 waves)

typedef __attribute__((ext_vector_type(2))) float v2f;
typedef __attribute__((ext_vector_type(8))) float v8f;

// ---------------------------------------------------------------
// Build fused DFT weights into workspace:
//   wc[j][n] = (-1)^j *  win[n] * cos(2*pi*j*n/1024)
//   ws[j][n] = (-1)^j * -win[n] * sin(2*pi*j*n/1024)
// win[n] = (0.54 - 0.46*cos(2*pi*n/1023)) / sqrt(1024)
// Argument reduced exactly via (j*n) mod 1024 so fp32 sincos stays accurate.
// ---------------------------------------------------------------
__global__ void stft_build_tables(float* __restrict__ wc, float* __restrict__ ws)
{
    int idx = blockIdx.x * blockDim.x + threadIdx.x;   // 0 .. NCOL_PAD*1024-1
    int j = idx >> 10;
    int n = idx & 1023;
    float win = (0.54f - 0.46f * cosf(6.2831853071795864769f * (float)n / 1023.0f))
                * 0.03125f;                            // 1/sqrt(1024) = 1/32
    int t = (j * n) & 1023;                            // exact mod-1024 reduction
    float th = (float)t * (6.2831853071795864769f / 1024.0f);
    float s, c;
    sincosf(th, &s, &c);
    float sign = (j & 1) ? -1.0f : 1.0f;
    wc[idx] = sign * win * c;
    ws[idx] = -sign * win * s;
}

// ---------------------------------------------------------------
// Zero the trailing rows [N_FRAMES, N_ROWS) of both mag and phase planes.
// ---------------------------------------------------------------
__global__ void stft_zero_tail(float* __restrict__ out)
{
    int i = blockIdx.x * blockDim.x + threadIdx.x;
    const int cnt = (N_ROWS - N_FRAMES) * HALF_N;      // 3 * 513
    if (i < cnt) {
        int row = N_FRAMES + i / HALF_N;
        int col = i % HALF_N;
        size_t o = (size_t)row * HALF_N + col;
        out[o] = 0.0f;
        out[(size_t)N_ROWS * HALF_N + o] = 0.0f;
    }
}

// ---------------------------------------------------------------
// Main STFT kernel. Block = 128 threads (4 waves).
//   blockIdx.x : frame tile (16 frames)
//   blockIdx.y : bin group (4 tiles of 16 bins, one per wave)
// LDS stages 16 frames x 1024 samples in [k][m] layout (conflict-free).
// Each wave runs a 1024-deep K loop of v_wmma_f32_16x16x4_f32, one
// accumulator for Re (cos weights) and one for Im (sin weights).
// ---------------------------------------------------------------
__global__ __launch_bounds__(128) void stft_wmma_kernel(
    const float* __restrict__ x,
    const float* __restrict__ wc,
    const float* __restrict__ wsn,
    float* __restrict__ out)
{
    __shared__ float As[N_FFT * 16];                   // 64 KB, [k][m]

    const int tid   = threadIdx.x;
    const int lane  = tid & 31;
    const int wave  = tid >> 5;
    const int fbase = blockIdx.x * 16;

    // ---- stage 16 overlapping frames into LDS (window folded into B) ----
    {
        int m = tid & 15;                              // frame within tile
        int chunk = tid >> 4;                          // 0..7
        int gbase = (fbase + m) * HOP - PAD;           // signal index of sample 0
        #pragma unroll 8
        for (int i = 0; i < 64; ++i) {
            int s  = chunk * 2 + i * 16;               // even sample offset
            int gi = gbase + s;                        // even -> pair never straddles
            float2 v;
            if ((unsigned)gi <= (unsigned)(SIG_LEN - 2))
                v = *(const float2*)(x + gi);
            else
                v = make_float2(0.0f, 0.0f);
            As[s * 16 + m]       = v.x;                // bank-conflict-free stores
            As[(s + 1) * 16 + m] = v.y;
        }
    }
    __syncthreads();

    // ---- per-wave WMMA tile: 16 frames x 16 bins ----
    const int m    = lane & 15;                        // A row (M) for this lane
    const int half = lane >> 4;                        // K-pair selector
    const int nb   = (blockIdx.y * 4 + wave) * 16 + m; // bin column (N) for this lane
    const float* bc = wc  + (size_t)nb * N_FFT;
    const float* bs = wsn + (size_t)nb * N_FFT;

    v8f cre = {};
    v8f cim = {};

    #pragma unroll 4
    for (int k0 = 0; k0 < N_FFT; k0 += 4) {
        int ka = k0 + 2 * half;                        // even -> 8B-aligned B loads
        v2f a;
        a.x = As[ka * 16 + m];                         // ISA f32 A 16x4 layout
        a.y = As[(ka + 1) * 16 + m];
        v2f b1 = *(const v2f*)(bc + ka);               // ISA f32 B 4x16 layout
        v2f b2 = *(const v2f*)(bs + ka);
        cre = __builtin_amdgcn_wmma_f32_16x16x4_f32(
                  false, a, false, b1, (short)0, cre, false, false);
        cim = __builtin_amdgcn_wmma_f32_16x16x4_f32(
                  false, a, false, b2, (short)0, cim, false, false);
    }

    // ---- epilogue: mag + phase, no cross-lane needed ----
    if (nb < HALF_N) {
        const int fb2 = fbase + 8 * half;              // C layout: lanes16-31 -> M+8
        #pragma unroll
        for (int r = 0; r < 8; ++r) {
            int fr = fb2 + r;
            if (fr < N_FRAMES) {
                float re  = cre[r];
                float im  = cim[r];
                float mag = sqrtf(re * re + im * im);
                float ph  = atan2f(im, re);
                size_t o  = (size_t)fr * HALF_N + nb;
                out[o] = mag;                          // xm_x plane
                out[(size_t)N_ROWS * HALF_N + o] = ph; // xp_x plane
            }
        }
    }
}

// ---------------------------------------------------------------
extern "C" void kernel_launch(void* const* d_in, const int* in_sizes, int n_in,
                              void* d_out, int out_size, void* d_ws, size_t ws_size,
                              hipStream_t stream)
{
    (void)in_sizes; (void)n_in; (void)out_size; (void)ws_size;

    const float* x = (const float*)d_in[0];
    float* out = (float*)d_out;

    float* wc = (float*)d_ws;                          // [NCOL_PAD][1024] cos weights
    float* ws = wc + (size_t)NCOL_PAD * N_FFT;         // [NCOL_PAD][1024] sin weights

    // 1) fused window+shift+DFT weight tables (recomputed every call: deterministic)
    stft_build_tables<<<(NCOL_PAD * N_FFT) / 256, 256, 0, stream>>>(wc, ws);

    // 2) zero the trailing (never-computed) rows of both output planes
    stft_zero_tail<<<((N_ROWS - N_FRAMES) * HALF_N + 255) / 256, 256, 0, stream>>>(out);

    // 3) STFT: 4097 frame tiles x 9 bin groups, 4 waves/block
    dim3 grid((N_FRAMES + 15) / 16, 9);
    stft_wmma_kernel<<<grid, 128, 0, stream>>>(x, wc, ws, out);
}